// ContrastiveLoss_40656160424043
// MI455X (gfx1250) — compile-verified
//
#include <hip/hip_runtime.h>
#include <hip/hip_bf16.h>

// CDNA5 / gfx1250, wave32.
typedef float v2f __attribute__((ext_vector_type(2)));
typedef float v8f __attribute__((ext_vector_type(8)));

static constexpr int   N_ROWS   = 8192;   // N == M == 8192
static constexpr int   DIM      = 768;
static constexpr int   KC       = 32;     // K chunk staged in LDS
static constexpr int   LDS_LD   = 36;     // padded stride: 144B rows -> 16B aligned, conflict-free b64 reads
static constexpr float MARGIN_F = 1.0f;
static constexpr float EPS_F    = 1e-6f;

// ---------------------------------------------------------------- init
__global__ void cl_init_acc(double* acc) {
    int t = threadIdx.x;
    if (t < 8) acc[t] = 0.0;
}

// ---------------------------------------------------------------- row stats: s = sum x^2, r = sum x
__global__ __launch_bounds__(256)
void cl_rowstats(const float* __restrict__ X, const float* __restrict__ Y,
                 float* __restrict__ sx, float* __restrict__ rx,
                 float* __restrict__ sy, float* __restrict__ ry) {
    const int lane = threadIdx.x & 31;
    const int wave = threadIdx.x >> 5;
    const int row  = blockIdx.x * 8 + wave;          // 0 .. 16383 (X rows then Y rows)
    const bool isX = row < N_ROWS;
    const int  r   = isX ? row : (row - N_ROWS);
    const float* __restrict__ P = (isX ? X : Y) + (size_t)r * DIM + lane;
    float s = 0.f, u = 0.f;
#pragma unroll
    for (int t = 0; t < DIM / 32; ++t) {
        float v = P[t * 32];
        s = fmaf(v, v, s);
        u += v;
    }
#pragma unroll
    for (int off = 16; off > 0; off >>= 1) {
        s += __shfl_down(s, off, 32);
        u += __shfl_down(u, off, 32);
    }
    if (lane == 0) {
        if (isX) { sx[r] = s; rx[r] = u; }
        else     { sy[r] = s; ry[r] = u; }
    }
}

// ---------------------------------------------------------------- main tile kernel
// MODE 0: jj pass (upper-triangle pos/neg losses, pos count)
// MODE 1: jn pass (cross loss, all pairs)
// Block = 256 threads = 8 waves. Block tile 128x128; wave tile 32x64 = 2x4 WMMA tiles.
// Software-pipelined: global loads for chunk k+1 issued before the WMMAs of chunk k,
// stored into the alternate LDS buffer after the WMMAs (double-buffered LDS).
template <int MODE>
__global__ __launch_bounds__(256)
void cl_tile(const float* __restrict__ A, const float* __restrict__ B,
             const float* __restrict__ sa, const float* __restrict__ ra,
             const float* __restrict__ sb, const float* __restrict__ rb,
             const long long* __restrict__ labels, double* __restrict__ acc) {
    const int bi = blockIdx.y, bj = blockIdx.x;
    const int i0 = bi * 128, j0 = bj * 128;
    // jj pass needs only i < j: fully-lower blocks contribute nothing.
    if (MODE == 0 && i0 >= j0 + 128) return;   // block-uniform exit (EXEC stays all-ones)

    __shared__ float sAt[2][128 * LDS_LD];
    __shared__ float sBt[2][128 * LDS_LD];
    __shared__ float red[8][3];

    const int tid  = threadIdx.x;
    const int lane = tid & 31;
    const int wave = tid >> 5;
    const int half = lane >> 4;   // K half-select for A/B fragments
    const int l16  = lane & 15;
    const int wm   = wave >> 1;   // 0..3 -> 32-row slice
    const int wn   = wave & 1;    // 0..1 -> 64-col slice

    // staging geometry (128 rows x 8 float4-cols, 4 units per thread)
    int srow[4], scol[4];
#pragma unroll
    for (int u = 0; u < 4; ++u) {
        const int idx = tid + u * 256;      // 0..1023
        srow[u] = idx >> 3;
        scol[u] = (idx & 7) * 4;
    }

    v8f c[2][4];
    const v8f vzero = {0.f, 0.f, 0.f, 0.f, 0.f, 0.f, 0.f, 0.f};
#pragma unroll
    for (int m = 0; m < 2; ++m)
#pragma unroll
        for (int n = 0; n < 4; ++n) c[m][n] = vzero;

    float4 rga[4], rgb[4];

    // prologue: load + store chunk 0
#pragma unroll
    for (int u = 0; u < 4; ++u) {
        rga[u] = *(const float4*)&A[(size_t)(i0 + srow[u]) * DIM + scol[u]];
        rgb[u] = *(const float4*)&B[(size_t)(j0 + srow[u]) * DIM + scol[u]];
    }
#pragma unroll
    for (int u = 0; u < 4; ++u) {
        *(float4*)&sAt[0][srow[u] * LDS_LD + scol[u]] = rga[u];
        *(float4*)&sBt[0][srow[u] * LDS_LD + scol[u]] = rgb[u];
    }
    __syncthreads();

    int p = 0;
    for (int kc = 0; kc < DIM; kc += KC) {
        const bool has_next = (kc + KC) < DIM;
        if (has_next) {
            const int kn = kc + KC;
#pragma unroll
            for (int u = 0; u < 4; ++u) {
                rga[u] = *(const float4*)&A[(size_t)(i0 + srow[u]) * DIM + kn + scol[u]];
                rgb[u] = *(const float4*)&B[(size_t)(j0 + srow[u]) * DIM + kn + scol[u]];
            }
        }
        // compute on buffer p while the loads above are in flight
        const float* __restrict__ pA = sAt[p];
        const float* __restrict__ pB = sBt[p];
#pragma unroll
        for (int kk = 0; kk < KC; kk += 4) {
            v2f af[2], bf[4];
#pragma unroll
            for (int m = 0; m < 2; ++m)
                af[m] = *(const v2f*)&pA[(wm * 32 + m * 16 + l16) * LDS_LD + kk + 2 * half];
#pragma unroll
            for (int n = 0; n < 4; ++n)
                bf[n] = *(const v2f*)&pB[(wn * 64 + n * 16 + l16) * LDS_LD + kk + 2 * half];
#pragma unroll
            for (int m = 0; m < 2; ++m)
#pragma unroll
                for (int n = 0; n < 4; ++n)
                    c[m][n] = __builtin_amdgcn_wmma_f32_16x16x4_f32(
                        false, af[m], false, bf[n], (short)0, c[m][n], false, false);
        }
        if (has_next) {
            const int q = p ^ 1;
#pragma unroll
            for (int u = 0; u < 4; ++u) {
                *(float4*)&sAt[q][srow[u] * LDS_LD + scol[u]] = rga[u];
                *(float4*)&sBt[q][srow[u] * LDS_LD + scol[u]] = rgb[u];
            }
        }
        __syncthreads();
        p ^= 1;
    }

    // ---------------- fused epilogue ----------------
    const float deps2 = (float)DIM * EPS_F * EPS_F;
    float jS[4], jR[4];
    long long jL[4];
    int jI[4];
#pragma unroll
    for (int n = 0; n < 4; ++n) {
        const int j = j0 + wn * 64 + n * 16 + l16;
        jI[n] = j; jS[n] = sb[j]; jR[n] = rb[j];
        jL[n] = (MODE == 0) ? labels[j] : 0ll;
    }

    float v0 = 0.f, v1 = 0.f, v2 = 0.f; // MODE0: pos_sum, neg_sum, pos_cnt ; MODE1: cross_sum
#pragma unroll
    for (int m = 0; m < 2; ++m) {
#pragma unroll
        for (int e = 0; e < 8; ++e) {
            const int i = i0 + wm * 32 + m * 16 + half * 8 + e; // C layout: M = e + 8*half
            const float si = sa[i], ri = ra[i];
            const long long li = (MODE == 0) ? labels[i] : 0ll;
#pragma unroll
            for (int n = 0; n < 4; ++n) {
                float d2 = si + jS[n] - 2.f * c[m][n][e]
                         + 2.f * EPS_F * (ri - jR[n]) + deps2;
                d2 = fmaxf(d2, 0.f);
                if (MODE == 0) {
                    if (i < jI[n]) {
                        if (li == jL[n]) { v0 += d2; v2 += 1.f; }
                        else {
                            float d = sqrtf(fmaxf(d2, 1e-12f));
                            float t = fmaxf(MARGIN_F - d, 0.f);
                            v1 = fmaf(t, t, v1);
                        }
                    }
                } else {
                    float d = sqrtf(fmaxf(d2, 1e-12f));
                    float t = fmaxf(MARGIN_F - d, 0.f);
                    v0 = fmaf(t, t, v0);
                }
            }
        }
    }

    // wave reduce (wave32), then block reduce, then double atomics
#pragma unroll
    for (int off = 16; off > 0; off >>= 1) {
        v0 += __shfl_down(v0, off, 32);
        v1 += __shfl_down(v1, off, 32);
        v2 += __shfl_down(v2, off, 32);
    }
    if (lane == 0) { red[wave][0] = v0; red[wave][1] = v1; red[wave][2] = v2; }
    __syncthreads();
    if (tid == 0) {
        float t0 = 0.f, t1 = 0.f, t2 = 0.f;
        for (int w = 0; w < 8; ++w) { t0 += red[w][0]; t1 += red[w][1]; t2 += red[w][2]; }
        if (MODE == 0) {
            atomicAdd(&acc[0], (double)t0);
            atomicAdd(&acc[1], (double)t1);
            atomicAdd(&acc[3], (double)t2);
        } else {
            atomicAdd(&acc[2], (double)t0);
        }
    }
}

// ---------------------------------------------------------------- finalize
__global__ void cl_finalize(const double* __restrict__ acc, float* __restrict__ out) {
    const double pos   = acc[0];
    const double neg   = acc[1];
    const double cross = acc[2];
    const double cnt   = acc[3];
    const double total_upper = (double)N_ROWS * (double)(N_ROWS - 1) * 0.5;
    const double n_pos = fmax(cnt, 1.0);
    const double n_neg = fmax(total_upper - cnt, 1.0);
    const double loss  = pos / n_pos + neg / n_neg
                       + cross / ((double)N_ROWS * (double)N_ROWS);
    out[0] = (float)loss;
}

// ---------------------------------------------------------------- launcher
extern "C" void kernel_launch(void* const* d_in, const int* in_sizes, int n_in,
                              void* d_out, int out_size, void* d_ws, size_t ws_size,
                              hipStream_t stream) {
    (void)in_sizes; (void)n_in; (void)out_size; (void)ws_size;
    const float*     X      = (const float*)d_in[0];      // [8192, 768]
    const float*     Y      = (const float*)d_in[1];      // [8192, 768]
    const long long* labels = (const long long*)d_in[2];  // [8192] int64
    float* out = (float*)d_out;

    char*   ws  = (char*)d_ws;
    double* acc = (double*)ws;                 // 8 doubles (uses 4)
    float*  sx  = (float*)(ws + 256);
    float*  rx  = sx + N_ROWS;
    float*  sy  = rx + N_ROWS;
    float*  ry  = sy + N_ROWS;                 // total scratch: 256 + 4*8192*4 B = ~128.3 KB

    cl_init_acc<<<1, 32, 0, stream>>>(acc);
    cl_rowstats<<<(2 * N_ROWS) / 8, 256, 0, stream>>>(X, Y, sx, rx, sy, ry);

    dim3 grid(N_ROWS / 128, N_ROWS / 128); // 64 x 64
    cl_tile<0><<<grid, 256, 0, stream>>>(X, X, sx, rx, sx, rx, labels, acc);
    cl_tile<1><<<grid, 256, 0, stream>>>(X, Y, sx, rx, sy, ry, labels, acc);

    cl_finalize<<<1, 1, 0, stream>>>(acc, out);
}